// MinibatchDiscriminator_64802466562626
// MI455X (gfx1250) — compile-verified
//
#include <hip/hip_runtime.h>

typedef __attribute__((ext_vector_type(2))) float v2f;
typedef __attribute__((ext_vector_type(8))) float v8f;

#define N_      512
#define INSIZE_ 512
#define K_      64
#define D_      16
#define KD_     1024   // K_*D_
#define OUTW_   576    // INSIZE_ + K_

// ---------------------------------------------------------------------------
// CDNA5 async global->LDS staging (ASYNCcnt-tracked), with safe fallback.
// Builtin signature (from clang diagnostic):
//   void __builtin_amdgcn_global_load_async_to_lds_b128(
//       int __vector(4) addrspace(1)* src, int __vector(4) addrspace(3)* dst,
//       imm int offset, imm int cpol)
// ---------------------------------------------------------------------------
#if defined(__has_builtin)
#if __has_builtin(__builtin_amdgcn_global_load_async_to_lds_b128)
#define HAVE_ASYNC_LDS 1
#endif
#endif

__device__ __forceinline__ void wait_async0() {
#if defined(__has_builtin) && __has_builtin(__builtin_amdgcn_s_wait_asynccnt)
  __builtin_amdgcn_s_wait_asynccnt(0);
#else
  asm volatile("s_wait_asynccnt 0x0" ::: "memory");
#endif
}

#ifdef HAVE_ASYNC_LDS
typedef int v4i_vs __attribute__((vector_size(16)));
typedef __attribute__((address_space(1))) v4i_vs* glb_b128_t;
typedef __attribute__((address_space(3))) v4i_vs* lds_b128_t;

__device__ __forceinline__ void async_copy_b128(const float* gp, float* sp) {
  __builtin_amdgcn_global_load_async_to_lds_b128(
      (glb_b128_t)(void*)gp, (lds_b128_t)sp, /*offset=*/0, /*cpol=*/0);
}
#endif

// -----------------------------------------------------------------------------
// feat[N_, KD_] = x[N_, INSIZE_] @ W[KD_, INSIZE_]^T + b
// One wave32 per 16x16 output tile; V_WMMA_F32_16X16X4_F32 over K in steps of 4.
//
// Lane layouts (CDNA5 ISA 7.12.2):
//   A (16x4 f32):  lane L -> M = L&15 ; VGPR{0,1} hold K = 2*(L>>4)+{0,1}
//   B (4x16 f32):  lane L -> N = L&15 ; VGPR{0,1} hold K = 2*(L>>4)+{0,1}
//   C/D (16x16 f32): VGPR v, lane L -> M = v + 8*(L>>4), N = L&15
// -----------------------------------------------------------------------------
__global__ __launch_bounds__(256) void gemm_wmma_f32(
    const float* __restrict__ x, const float* __restrict__ W,
    const float* __restrict__ bias, float* __restrict__ feat) {
  const int wave = (blockIdx.x * blockDim.x + threadIdx.x) >> 5;
  const int lane = threadIdx.x & 31;

  const int tiles_n = KD_ / 16;                 // 64
  const int mt = wave / tiles_n;                // 0..31
  const int nt = wave % tiles_n;                // 0..63
  const int m0 = mt * 16, n0 = nt * 16;

  const int half = lane >> 4;                   // 0 or 1
  const int r    = lane & 15;

  const float* __restrict__ xrow = x + (size_t)(m0 + r) * INSIZE_;
  const float* __restrict__ wrow = W + (size_t)(n0 + r) * INSIZE_;

  // C init: every D element at column n0 + (lane&15) gets bias[n0 + r]
  const float bv = bias[n0 + r];
  v8f c;
#pragma unroll
  for (int v = 0; v < 8; ++v) c[v] = bv;

#pragma unroll 8
  for (int kk = 0; kk < INSIZE_; kk += 4) {
    v2f a  = *(const v2f*)(xrow + kk + half * 2);   // A[M][kk + 2*half + {0,1}]
    v2f bb = *(const v2f*)(wrow + kk + half * 2);   // B[k][N] = W[n0+r][kk + ...]
    c = __builtin_amdgcn_wmma_f32_16x16x4_f32(
        /*neg_a=*/false, a, /*neg_b=*/false, bb,
        /*c_mod=*/(short)0, c, /*reuse_a=*/false, /*reuse_b=*/false);
  }

#pragma unroll
  for (int v = 0; v < 8; ++v) {
    feat[(size_t)(m0 + v + 8 * half) * KD_ + (n0 + r)] = c[v];
  }
}

// -----------------------------------------------------------------------------
// Grid (K_=64, 4). Block (k, iq): stage feat[:, k*16 .. k*16+16) (32 KB) into
// LDS via async-to-LDS, then compute o_b[i,k] for i in [iq*128, iq*128+128).
// j-range split across thread halves (tid>>7), combined via LDS partials.
// All lanes iterate j in lockstep -> LDS reads are wave-broadcast.
// -----------------------------------------------------------------------------
__global__ __launch_bounds__(256) void l1_exp_reduce(
    const float* __restrict__ feat, float* __restrict__ out) {
  __shared__ float sF[N_][D_];                  // 512*16*4 = 32 KB
  __shared__ float partial[256];
  const int k   = blockIdx.x;
  const int iq  = blockIdx.y;                   // 0..3
  const int tid = threadIdx.x;

  // stage all 512 rows x 16 floats: 2048 b128 transfers, 8 per thread
#ifdef HAVE_ASYNC_LDS
  for (int t = tid; t < N_ * (D_ / 4); t += 256) {
    const int i = t >> 2, q = t & 3;
    async_copy_b128(feat + (size_t)i * KD_ + k * D_ + q * 4, &sF[i][q * 4]);
  }
  wait_async0();
  __syncthreads();
#else
  for (int t = tid; t < N_ * (D_ / 4); t += 256) {
    const int i = t >> 2, q = t & 3;
    float4 v = ((const float4*)(feat + (size_t)i * KD_ + k * D_))[q];
    ((float4*)&sF[i][0])[q] = v;
  }
  __syncthreads();
#endif

  const int i  = iq * 128 + (tid & 127);        // row this thread scores
  const int jh = tid >> 7;                      // 0 -> j in [0,256), 1 -> [256,512)

  float fi[D_];
#pragma unroll
  for (int d = 0; d < D_; ++d) fi[d] = sF[i][d];

  float acc = 0.0f;
  const int j0 = jh * (N_ / 2), j1 = j0 + (N_ / 2);
  for (int j = j0; j < j1; ++j) {
    float l1 = 0.0f;
#pragma unroll
    for (int d = 0; d < D_; ++d) l1 += fabsf(fi[d] - sF[j][d]);
    // diagonal: norm + 1e6 penalty -> exp ~ 0, so skip j == i
    acc += (j == i) ? 0.0f : __expf(-l1);
  }
  partial[tid] = acc;
  __syncthreads();

  if (tid < 128) {
    out[(size_t)i * OUTW_ + INSIZE_ + k] = partial[tid] + partial[tid + 128];
  }
}

// -----------------------------------------------------------------------------
// out[i, 0:512] = x[i, :]  (float4 copy; 512 rows * 128 float4s)
// -----------------------------------------------------------------------------
__global__ __launch_bounds__(256) void copy_x(
    const float* __restrict__ x, float* __restrict__ out) {
  const int idx = blockIdx.x * blockDim.x + threadIdx.x;  // 0 .. 512*128-1
  const int i = idx >> 7;
  const int q = idx & 127;
  float4 v = ((const float4*)(x + (size_t)i * INSIZE_))[q];
  ((float4*)(out + (size_t)i * OUTW_))[q] = v;
}

extern "C" void kernel_launch(void* const* d_in, const int* in_sizes, int n_in,
                              void* d_out, int out_size, void* d_ws, size_t ws_size,
                              hipStream_t stream) {
  const float* x = (const float*)d_in[0];   // [512, 512]
  const float* W = (const float*)d_in[1];   // [1024, 512]
  const float* b = (const float*)d_in[2];   // [1024]
  float* out  = (float*)d_out;              // [512, 576]
  float* feat = (float*)d_ws;               // [512, 1024] = 2 MB scratch

  // 2048 tiles / 8 waves per 256-thread block = 256 blocks
  gemm_wmma_f32<<<256, 256, 0, stream>>>(x, W, b, feat);
  copy_x<<<256, 256, 0, stream>>>(x, out);
  l1_exp_reduce<<<dim3(K_, 4), 256, 0, stream>>>(feat, out);
}